// GCN_10376640987777
// MI455X (gfx1250) — compile-verified
//
#include <hip/hip_runtime.h>
#include <hip/hip_bf16.h>

typedef __attribute__((ext_vector_type(2))) float v2f;
typedef __attribute__((ext_vector_type(8))) float v8f;

#define HID 64

// ---------------------------------------------------------------------------
// Degree: deg[col] += 1 per edge (self-loop +1 folded into dinv kernel)
// ---------------------------------------------------------------------------
__global__ void gcn_degree_kernel(const int* __restrict__ ei, float* __restrict__ deg, int E) {
    int e = blockIdx.x * blockDim.x + threadIdx.x;
    if (e < E) atomicAdd(&deg[ei[E + e]], 1.0f);
}

__global__ void gcn_dinv_kernel(const float* __restrict__ deg, float* __restrict__ dinv, int N) {
    int i = blockIdx.x * blockDim.x + threadIdx.x;
    if (i < N) dinv[i] = rsqrtf(deg[i] + 1.0f);   // +1 = self loop; always > 0
}

// ---------------------------------------------------------------------------
// s = dinv[:,None] * (h @ W)  via V_WMMA_F32_16X16X4_F32, LDS-staged operands.
// Block: 256 threads = 8 waves. Block tile: 32 node-rows x 64 feature-cols.
// Each wave owns one 16x16 output tile (2 row subtiles x 4 col subtiles).
// LDS H-tile row stride padded to K+4 floats: 16B aligned float4 stores and
// conflict-free v2f reads (bank = 4*m + ka, distinct per half-wave lane).
// A frag (16x4 f32): lane m=lane&15 is row M; kh=lane>>4 selects K pair.
// B frag (4x16 f32): lane m is col N; kh selects K pair (symmetric layout).
// D frag: 8 VGPRs, M = i + 8*kh, N = m.
// ---------------------------------------------------------------------------
template <int K>
__global__ void gcn_gemm_scale_wmma(const float* __restrict__ h,
                                    const float* __restrict__ W,
                                    const float* __restrict__ dinv,
                                    float* __restrict__ s, int N) {
    constexpr int HS = K + 4;                // padded LDS row stride (floats)
    __shared__ float lds_h[32 * HS];
    __shared__ float lds_W[K * HID];
    __shared__ float lds_d[32];

    const int tid      = threadIdx.x;
    const int rowBase0 = blockIdx.x * 32;

    // ---- stage H tile: 32 x K floats, contiguous in global -> coalesced ----
    constexpr int HQ = 32 * K / 4;           // float4 count
#pragma unroll
    for (int idx = tid; idx < HQ; idx += 256) {
        const int r  = idx / (K / 4);
        const int kq = idx % (K / 4);
        const long long gr = min(rowBase0 + r, N - 1);
        const float4 v = *(const float4*)(h + gr * K + kq * 4);
        *(float4*)(lds_h + r * HS + kq * 4) = v;
    }
    // ---- stage W: K x 64 floats, contiguous ----
    constexpr int WQ = K * HID / 4;
#pragma unroll
    for (int idx = tid; idx < WQ; idx += 256) {
        *(float4*)(lds_W + idx * 4) = *(const float4*)(W + idx * 4);
    }
    if (tid < 32) lds_d[tid] = dinv[min(rowBase0 + tid, N - 1)];
    __syncthreads();

    const int lane    = tid & 31;
    const int wave    = tid >> 5;                 // 0..7
    const int rowSub  = (wave >> 2) * 16;         // 0 or 16 within block tile
    const int colBase = (wave & 3) * 16;          // 4 col tiles of 16
    const int m       = lane & 15;
    const int kh      = lane >> 4;                // 0 or 1

    v8f c = {};
#pragma unroll
    for (int k0 = 0; k0 < K; k0 += 4) {
        const int ka = k0 + 2 * kh;
        const v2f a = *(const v2f*)(lds_h + (rowSub + m) * HS + ka);
        v2f b;
        b.x = lds_W[ka * HID + colBase + m];
        b.y = lds_W[(ka + 1) * HID + colBase + m];
        c = __builtin_amdgcn_wmma_f32_16x16x4_f32(false, a, false, b,
                                                  (short)0, c, false, false);
    }

    if (rowBase0 + 32 <= N) {                     // full tile: unguarded stores
#pragma unroll
        for (int i = 0; i < 8; ++i) {
            const int rl = rowSub + 8 * kh + i;
            s[(long long)(rowBase0 + rl) * HID + colBase + m] = c[i] * lds_d[rl];
        }
    } else {
#pragma unroll
        for (int i = 0; i < 8; ++i) {
            const int rl = rowSub + 8 * kh + i;
            if (rowBase0 + rl < N)
                s[(long long)(rowBase0 + rl) * HID + colBase + m] = c[i] * lds_d[rl];
        }
    }
}

// ---------------------------------------------------------------------------
// Edge scatter: acc[col] += s[row]. 16 threads per edge, float4 gathers,
// fp32 global atomics (global_atomic_add_f32).
// ---------------------------------------------------------------------------
__global__ void gcn_scatter_kernel(const int* __restrict__ ei,
                                   const float* __restrict__ s,
                                   float* __restrict__ acc, int E) {
    int t = blockIdx.x * blockDim.x + threadIdx.x;   // E*16 <= 25.6M, fits int
    int e = t >> 4;
    if (e >= E) return;
    int f = (t & 15) * 4;
    int r = ei[e];           // source
    int c = ei[E + e];       // target
    const float4 v = *(const float4*)(s + (long long)r * HID + f);
    float* a = acc + (long long)c * HID + f;
    atomicAdd(a + 0, v.x);
    atomicAdd(a + 1, v.y);
    atomicAdd(a + 2, v.z);
    atomicAdd(a + 3, v.w);
}

// ---------------------------------------------------------------------------
// hout = act( dinv[i] * (acc + s) + b )   (self-loop folded in as +s)
// ---------------------------------------------------------------------------
__global__ void gcn_finalize_kernel(const float4* __restrict__ accv,
                                    const float4* __restrict__ sv,
                                    const float* __restrict__ dinv,
                                    const float* __restrict__ b,
                                    float4* __restrict__ outv, int N, int relu) {
    int t = blockIdx.x * blockDim.x + threadIdx.x;   // N * 16
    if (t >= N * 16) return;
    int i  = t >> 4;
    int fq = (t & 15) * 4;
    float di = dinv[i];
    float4 a = accv[t], sq = sv[t], r;
    r.x = di * (a.x + sq.x) + b[fq + 0];
    r.y = di * (a.y + sq.y) + b[fq + 1];
    r.z = di * (a.z + sq.z) + b[fq + 2];
    r.w = di * (a.w + sq.w) + b[fq + 3];
    if (relu) {
        r.x = fmaxf(r.x, 0.0f); r.y = fmaxf(r.y, 0.0f);
        r.z = fmaxf(r.z, 0.0f); r.w = fmaxf(r.w, 0.0f);
    }
    outv[t] = r;
}

// ---------------------------------------------------------------------------
// Mean-pool accumulation: sums[batch[i]] += h[i]; cnts[batch[i]] += 1
// ---------------------------------------------------------------------------
__global__ void gcn_pool_kernel(const float* __restrict__ h,
                                const int* __restrict__ batch,
                                float* __restrict__ sums,
                                float* __restrict__ cnts, int N) {
    int t = blockIdx.x * blockDim.x + threadIdx.x;   // N * 16
    if (t >= N * 16) return;
    int i = t >> 4;
    int f = (t & 15) * 4;
    int g = batch[i];
    const float4 v = *(const float4*)(h + (long long)i * HID + f);
    float* a = sums + g * HID + f;
    atomicAdd(a + 0, v.x);
    atomicAdd(a + 1, v.y);
    atomicAdd(a + 2, v.z);
    atomicAdd(a + 3, v.w);
    if (f == 0) atomicAdd(&cnts[g], 1.0f);
}

// ---------------------------------------------------------------------------
// Head: out[g] = (sums[g]/max(cnt,1)) @ Wl + bl   (256 graphs x 2 classes)
// ---------------------------------------------------------------------------
__global__ void gcn_head_kernel(const float* __restrict__ sums,
                                const float* __restrict__ cnts,
                                const float* __restrict__ Wl,
                                const float* __restrict__ bl,
                                float* __restrict__ out, int G) {
    int g = blockIdx.x * blockDim.x + threadIdx.x;
    if (g >= G) return;
    float inv = 1.0f / fmaxf(cnts[g], 1.0f);
    float c0 = bl[0], c1 = bl[1];
#pragma unroll
    for (int k = 0; k < HID; ++k) {
        float p = sums[g * HID + k] * inv;
        c0 += p * Wl[k * 2 + 0];
        c1 += p * Wl[k * 2 + 1];
    }
    out[g * 2 + 0] = c0;
    out[g * 2 + 1] = c1;
}

// ---------------------------------------------------------------------------
extern "C" void kernel_launch(void* const* d_in, const int* in_sizes, int n_in,
                              void* d_out, int out_size, void* d_ws, size_t ws_size,
                              hipStream_t stream) {
    const float* x   = (const float*)d_in[0];
    const int*   ei  = (const int*)d_in[1];
    const int*   bat = (const int*)d_in[2];
    const float* W1  = (const float*)d_in[3];
    const float* b1  = (const float*)d_in[4];
    const float* W2  = (const float*)d_in[5];
    const float* b2  = (const float*)d_in[6];
    const float* W3  = (const float*)d_in[7];
    const float* b3  = (const float*)d_in[8];
    const float* Wl  = (const float*)d_in[9];
    const float* bl  = (const float*)d_in[10];
    float* out = (float*)d_out;

    const int IN_F = 32;
    const int N = in_sizes[0] / IN_F;     // 100000
    const int E = in_sizes[1] / 2;        // 1600000
    const int G = out_size / 2;           // 256

    // Workspace layout (floats)
    float* p = (float*)d_ws;
    float* deg  = p;  p += N;
    float* dinv = p;  p += N;
    float* sbuf = p;  p += (size_t)N * HID;
    float* acc  = p;  p += (size_t)N * HID;
    float* hbuf = p;  p += (size_t)N * HID;
    float* sums = p;  p += (size_t)G * HID;
    float* cnts = p;  p += G;
    (void)ws_size; (void)n_in;

    const int T = 256;
    const size_t featBytes = (size_t)N * HID * sizeof(float);

    // --- normalization ---
    hipMemsetAsync(deg, 0, (size_t)N * sizeof(float), stream);
    gcn_degree_kernel<<<(E + T - 1) / T, T, 0, stream>>>(ei, deg, E);
    gcn_dinv_kernel<<<(N + T - 1) / T, T, 0, stream>>>(deg, dinv, N);

    const int gemmBlocks = (N + 31) / 32;
    const int edgeBlocks = (E * 16 + T - 1) / T;
    const int featBlocks = (N * 16 + T - 1) / T;

    // --- layer 1: x[N,32] -> hbuf[N,64], relu ---
    gcn_gemm_scale_wmma<32><<<gemmBlocks, T, 0, stream>>>(x, W1, dinv, sbuf, N);
    hipMemsetAsync(acc, 0, featBytes, stream);
    gcn_scatter_kernel<<<edgeBlocks, T, 0, stream>>>(ei, sbuf, acc, E);
    gcn_finalize_kernel<<<featBlocks, T, 0, stream>>>((const float4*)acc, (const float4*)sbuf,
                                                      dinv, b1, (float4*)hbuf, N, 1);

    // --- layer 2: hbuf -> hbuf, relu ---
    gcn_gemm_scale_wmma<64><<<gemmBlocks, T, 0, stream>>>(hbuf, W2, dinv, sbuf, N);
    hipMemsetAsync(acc, 0, featBytes, stream);
    gcn_scatter_kernel<<<edgeBlocks, T, 0, stream>>>(ei, sbuf, acc, E);
    gcn_finalize_kernel<<<featBlocks, T, 0, stream>>>((const float4*)acc, (const float4*)sbuf,
                                                      dinv, b2, (float4*)hbuf, N, 1);

    // --- layer 3: hbuf -> hbuf, no relu ---
    gcn_gemm_scale_wmma<64><<<gemmBlocks, T, 0, stream>>>(hbuf, W3, dinv, sbuf, N);
    hipMemsetAsync(acc, 0, featBytes, stream);
    gcn_scatter_kernel<<<edgeBlocks, T, 0, stream>>>(ei, sbuf, acc, E);
    gcn_finalize_kernel<<<featBlocks, T, 0, stream>>>((const float4*)acc, (const float4*)sbuf,
                                                      dinv, b3, (float4*)hbuf, N, 0);

    // --- global mean pool + linear head ---
    hipMemsetAsync(sums, 0, (size_t)G * HID * sizeof(float), stream);
    hipMemsetAsync(cnts, 0, (size_t)G * sizeof(float), stream);
    gcn_pool_kernel<<<featBlocks, T, 0, stream>>>(hbuf, bat, sums, cnts, N);
    gcn_head_kernel<<<(G + T - 1) / T, T, 0, stream>>>(sums, cnts, Wl, bl, out, G);
}